// ResidualVQ_13898514170570
// MI455X (gfx1250) — compile-verified
//
#include <hip/hip_runtime.h>
#include <cstdint>

#define N_BATCH 32
#define C_DIM   512
#define T_DIM   1024
#define NT      (N_BATCH * T_DIM)   // 32768 rows
#define QNUM    6
#define KCODES  1024
#define UPD_BLOCKS (NT / 8)         // 4096 loss partials per layer

typedef float v2f __attribute__((ext_vector_type(2)));
typedef float v8f __attribute__((ext_vector_type(8)));

// Order-preserving float -> uint mapping (handles negatives; NaN-free data).
__device__ __forceinline__ unsigned fkey(float f) {
  unsigned u = __float_as_uint(f);
  return (u & 0x80000000u) ? ~u : (u | 0x80000000u);
}

__device__ __forceinline__ unsigned long long shfl_xor_u64(unsigned long long v, int m) {
  unsigned lo = __shfl_xor((unsigned)(v & 0xffffffffull), m, 32);
  unsigned hi = __shfl_xor((unsigned)(v >> 32), m, 32);
  return ((unsigned long long)hi << 32) | (unsigned long long)lo;
}

// ---------------------------------------------------------------------------
// x (N, C, T) -> residual (N*T, C)   [tiled transpose]
// ---------------------------------------------------------------------------
__global__ void k_transpose_in(const float* __restrict__ x, float* __restrict__ resid) {
  __shared__ float tile[32][33];
  const int n  = blockIdx.z;
  const int t0 = blockIdx.x * 32;
  const int c0 = blockIdx.y * 32;
  const float* xb = x + (size_t)n * C_DIM * T_DIM;
  for (int i = threadIdx.y; i < 32; i += 8)
    tile[i][threadIdx.x] = xb[(size_t)(c0 + i) * T_DIM + t0 + threadIdx.x]; // coalesced in t
  __syncthreads();
  float* rb = resid + ((size_t)n * T_DIM + t0) * C_DIM + c0;
  for (int i = threadIdx.y; i < 32; i += 8)
    rb[(size_t)i * C_DIM + threadIdx.x] = tile[threadIdx.x][i];            // coalesced in c
}

// ---------------------------------------------------------------------------
// cnorm[q*K + k] = sum_c cb[q][k][c]^2    (one wave per code row)
// ---------------------------------------------------------------------------
__global__ void k_cnorm(const float* __restrict__ cb, float* __restrict__ cnorm) {
  const int row  = (blockIdx.x * blockDim.x + threadIdx.x) >> 5;
  const int lane = threadIdx.x & 31;
  if (row >= QNUM * KCODES) return;
  const float* r = cb + (size_t)row * C_DIM;
  float s = 0.f;
  for (int c = lane * 4; c < C_DIM; c += 32 * 4) {
    float4 v = *(const float4*)(r + c);
    s += v.x * v.x + v.y * v.y + v.z * v.z + v.w * v.w;
  }
  for (int m = 16; m >= 1; m >>= 1) s += __shfl_xor(s, m, 32);
  if (lane == 0) cnorm[row] = s;
}

__global__ void k_zero_counts(int* __restrict__ counts) {
  int i = blockIdx.x * 256 + threadIdx.x;
  if (i < QNUM * KCODES) counts[i] = 0;
}

__global__ void k_init_best(unsigned long long* __restrict__ best) {
  int i = blockIdx.x * 256 + threadIdx.x;
  if (i < NT) best[i] = ~0ull;
}

// ---------------------------------------------------------------------------
// WMMA argmin.
// Block = 256 rows x 64 codes, 8 waves; wave = 2 M-tiles (32 rows) x 4 N-tiles.
// Inner loop: 6 fragment loads per 8 wmma, software-pipelined 2 deep so
// global loads overlap the wmma stream instead of serializing on loadcnt 0.
// dist_key(row, code) = cnorm[code] - 2 * <residual_row, cb_code>  (rnorm const)
// best[row] = min over codes of (fkey(dist) << 32 | code) via atomicMin u64.
// ---------------------------------------------------------------------------
struct Frag { v2f a0, a1, b0, b1, b2, b3; };

__device__ __forceinline__ Frag ldfrag(const float* __restrict__ Ap0,
                                       const float* __restrict__ Ap1,
                                       const float* __restrict__ Bp0,
                                       const float* __restrict__ Bp1,
                                       const float* __restrict__ Bp2,
                                       const float* __restrict__ Bp3, int c0) {
  Frag f;
  f.a0 = *(const v2f*)(Ap0 + c0);
  f.a1 = *(const v2f*)(Ap1 + c0);
  f.b0 = *(const v2f*)(Bp0 + c0);
  f.b1 = *(const v2f*)(Bp1 + c0);
  f.b2 = *(const v2f*)(Bp2 + c0);
  f.b3 = *(const v2f*)(Bp3 + c0);
  return f;
}

__device__ __forceinline__ void dofma(v8f acc[2][4], const Frag& f) {
  acc[0][0] = __builtin_amdgcn_wmma_f32_16x16x4_f32(false, f.a0, false, f.b0, (short)0, acc[0][0], false, false);
  acc[0][1] = __builtin_amdgcn_wmma_f32_16x16x4_f32(false, f.a0, false, f.b1, (short)0, acc[0][1], false, false);
  acc[0][2] = __builtin_amdgcn_wmma_f32_16x16x4_f32(false, f.a0, false, f.b2, (short)0, acc[0][2], false, false);
  acc[0][3] = __builtin_amdgcn_wmma_f32_16x16x4_f32(false, f.a0, false, f.b3, (short)0, acc[0][3], false, false);
  acc[1][0] = __builtin_amdgcn_wmma_f32_16x16x4_f32(false, f.a1, false, f.b0, (short)0, acc[1][0], false, false);
  acc[1][1] = __builtin_amdgcn_wmma_f32_16x16x4_f32(false, f.a1, false, f.b1, (short)0, acc[1][1], false, false);
  acc[1][2] = __builtin_amdgcn_wmma_f32_16x16x4_f32(false, f.a1, false, f.b2, (short)0, acc[1][2], false, false);
  acc[1][3] = __builtin_amdgcn_wmma_f32_16x16x4_f32(false, f.a1, false, f.b3, (short)0, acc[1][3], false, false);
}

__global__ void __launch_bounds__(256) k_argmin(
    const float* __restrict__ resid,
    const float* __restrict__ cbL,     // layer base (KCODES, C_DIM)
    const float* __restrict__ cnormL,  // layer base (KCODES)
    unsigned long long* __restrict__ best) {
  const int wave = threadIdx.x >> 5;
  const int lane = threadIdx.x & 31;
  const int half = lane >> 4;          // 0: lanes 0-15, 1: lanes 16-31
  const int sub  = lane & 15;
  const int rbase = blockIdx.x * 256 + wave * 32;
  const int cbase = blockIdx.y * 64;

  // A fragment bases: rows (rbase+sub), (rbase+16+sub); cols c0 + 2*half + {0,1}
  const float* Ap0 = resid + (size_t)(rbase + sub) * C_DIM + 2 * half;
  const float* Ap1 = Ap0 + (size_t)16 * C_DIM;
  // B fragment bases: codes (cbase + 16t + sub)
  const float* Bp0 = cbL + (size_t)(cbase + sub) * C_DIM + 2 * half;
  const float* Bp1 = Bp0 + (size_t)16 * C_DIM;
  const float* Bp2 = Bp0 + (size_t)32 * C_DIM;
  const float* Bp3 = Bp0 + (size_t)48 * C_DIM;

  float cn[4];
#pragma unroll
  for (int t = 0; t < 4; ++t) cn[t] = cnormL[cbase + 16 * t + sub];

  v8f acc[2][4];
#pragma unroll
  for (int m = 0; m < 2; ++m)
#pragma unroll
    for (int t = 0; t < 4; ++t) acc[m][t] = v8f{};

  // software pipeline: two stages in flight
  Frag f0 = ldfrag(Ap0, Ap1, Bp0, Bp1, Bp2, Bp3, 0);
  Frag f1 = ldfrag(Ap0, Ap1, Bp0, Bp1, Bp2, Bp3, 4);
  for (int c0 = 0; c0 < C_DIM - 8; c0 += 8) {
    Frag n0 = ldfrag(Ap0, Ap1, Bp0, Bp1, Bp2, Bp3, c0 + 8);
    dofma(acc, f0);
    Frag n1 = ldfrag(Ap0, Ap1, Bp0, Bp1, Bp2, Bp3, c0 + 12);
    dofma(acc, f1);
    f0 = n0;
    f1 = n1;
  }
  dofma(acc, f0);   // c0 = 504
  dofma(acc, f1);   // c0 = 508

  // acc[m][t][j] at this lane = cross[rbase + 16m + j + 8*half][cbase + 16t + sub]
#pragma unroll
  for (int m = 0; m < 2; ++m) {
#pragma unroll
    for (int j = 0; j < 8; ++j) {
      unsigned long long bp = ~0ull;
#pragma unroll
      for (int t = 0; t < 4; ++t) {
        const int code = cbase + 16 * t + sub;
        const float dist = cn[t] - 2.0f * acc[m][t][j];
        unsigned long long p = ((unsigned long long)fkey(dist) << 32) | (unsigned)code;
        bp = (p < bp) ? p : bp;
      }
      // reduce across the 16 lanes of this half (xor 1,2,4,8 stays within the half)
      for (int s = 1; s < 16; s <<= 1) {
        unsigned long long o = shfl_xor_u64(bp, s);
        bp = (o < bp) ? o : bp;
      }
      if (sub == 0) atomicMin(&best[rbase + 16 * m + j + 8 * half], bp);
    }
  }
}

// ---------------------------------------------------------------------------
// residual -= cb[code]; histogram; per-block loss partial (deterministic).
// One wave per row, 8 rows per block.
// ---------------------------------------------------------------------------
__global__ void k_update(float* __restrict__ resid,
                         const float* __restrict__ cbL,
                         const unsigned long long* __restrict__ best,
                         int* __restrict__ countsL,
                         float* __restrict__ lpartL) {
  __shared__ float ws[8];
  const int row  = (blockIdx.x * blockDim.x + threadIdx.x) >> 5;
  const int wave = threadIdx.x >> 5;
  const int lane = threadIdx.x & 31;
  const int code = (int)(best[row] & 0xffffffffull);
  if (lane == 0) atomicAdd(&countsL[code], 1);

  float* r = resid + (size_t)row * C_DIM;
  const float* e = cbL + (size_t)code * C_DIM;
  float s = 0.f;
  for (int c = lane * 4; c < C_DIM; c += 32 * 4) {
    float4 rv = *(const float4*)(r + c);
    float4 ev = *(const float4*)(e + c);
    rv.x -= ev.x; rv.y -= ev.y; rv.z -= ev.z; rv.w -= ev.w;
    s += rv.x * rv.x + rv.y * rv.y + rv.z * rv.z + rv.w * rv.w;
    *(float4*)(r + c) = rv;
  }
  for (int m = 16; m >= 1; m >>= 1) s += __shfl_xor(s, m, 32);
  if (lane == 0) ws[wave] = s;
  __syncthreads();
  if (threadIdx.x == 0) {
    float t = 0.f;
    for (int w = 0; w < 8; ++w) t += ws[w];
    lpartL[blockIdx.x] = t;
  }
}

// ---------------------------------------------------------------------------
// Scalars: all_losses, all_perplexity  (single block, deterministic trees)
// ---------------------------------------------------------------------------
__global__ void k_finalize(const int* __restrict__ counts,
                           const float* __restrict__ lpart,
                           float* __restrict__ out_scalars) {
  __shared__ float red[256];
  const int tid = threadIdx.x;
  float perp_sum = 0.f, loss_sum = 0.f;
  for (int q = 0; q < QNUM; ++q) {
    float h = 0.f;
    for (int k = tid; k < KCODES; k += 256) {
      float p = (float)counts[q * KCODES + k] * (1.0f / (float)NT);
      h += p * logf(p + 1e-7f);
    }
    red[tid] = h; __syncthreads();
    for (int s = 128; s > 0; s >>= 1) { if (tid < s) red[tid] += red[tid + s]; __syncthreads(); }
    if (tid == 0) perp_sum += expf(-red[0]);
    __syncthreads();

    float l = 0.f;
    for (int b = tid; b < UPD_BLOCKS; b += 256) l += lpart[q * UPD_BLOCKS + b];
    red[tid] = l; __syncthreads();
    for (int s = 128; s > 0; s >>= 1) { if (tid < s) red[tid] += red[tid + s]; __syncthreads(); }
    if (tid == 0) loss_sum += red[0];
    __syncthreads();
  }
  if (tid == 0) {
    out_scalars[0] = loss_sum / ((float)QNUM * (float)NT * (float)C_DIM);
    out_scalars[1] = perp_sum / (float)QNUM;
  }
}

// ---------------------------------------------------------------------------
// out[n][c][t] = x[n][c][t] - residual_final[(n*T+t)][c]   [tiled transpose]
// ---------------------------------------------------------------------------
__global__ void k_output(const float* __restrict__ x,
                         const float* __restrict__ resid,
                         float* __restrict__ out) {
  __shared__ float tile[32][33];
  const int n  = blockIdx.z;
  const int t0 = blockIdx.x * 32;
  const int c0 = blockIdx.y * 32;
  const float* rb = resid + ((size_t)n * T_DIM + t0) * C_DIM + c0;
  for (int i = threadIdx.y; i < 32; i += 8)
    tile[i][threadIdx.x] = rb[(size_t)i * C_DIM + threadIdx.x]; // tile[t_loc][c_loc]
  __syncthreads();
  const float* xb = x + (size_t)n * C_DIM * T_DIM;
  float* ob = out + (size_t)n * C_DIM * T_DIM;
  for (int i = threadIdx.y; i < 32; i += 8) {
    const size_t off = (size_t)(c0 + i) * T_DIM + t0 + threadIdx.x;
    ob[off] = xb[off] - tile[threadIdx.x][i];
  }
}

// ---------------------------------------------------------------------------
extern "C" void kernel_launch(void* const* d_in, const int* in_sizes, int n_in,
                              void* d_out, int out_size, void* d_ws, size_t ws_size,
                              hipStream_t stream) {
  const float* x   = (const float*)d_in[0];  // (32, 512, 1024)
  const float* cbs = (const float*)d_in[1];  // (6, 1024, 512)
  float* out = (float*)d_out;                // 32*512*1024 + 2 floats

  // workspace carve
  char* w = (char*)d_ws;
  float* resid = (float*)w;                          w += (size_t)NT * C_DIM * 4;        // 64 MB
  unsigned long long* best = (unsigned long long*)w; w += (size_t)NT * 8;                // 256 KB
  int*   counts = (int*)w;                           w += (size_t)QNUM * KCODES * 4;     // 24 KB
  float* cnorm  = (float*)w;                         w += (size_t)QNUM * KCODES * 4;     // 24 KB
  float* lpart  = (float*)w;                         w += (size_t)QNUM * UPD_BLOCKS * 4; // 96 KB
  (void)ws_size; (void)in_sizes; (void)n_in; (void)out_size;

  k_transpose_in<<<dim3(T_DIM / 32, C_DIM / 32, N_BATCH), dim3(32, 8), 0, stream>>>(x, resid);
  k_cnorm<<<(QNUM * KCODES * 32 + 255) / 256, 256, 0, stream>>>(cbs, cnorm);
  k_zero_counts<<<(QNUM * KCODES + 255) / 256, 256, 0, stream>>>(counts);

  for (int q = 0; q < QNUM; ++q) {
    const float* cbL = cbs + (size_t)q * KCODES * C_DIM;
    k_init_best<<<NT / 256, 256, 0, stream>>>(best);
    k_argmin<<<dim3(NT / 256, KCODES / 64), 256, 0, stream>>>(resid, cbL, cnorm + q * KCODES, best);
    k_update<<<NT / 8, 256, 0, stream>>>(resid, cbL, best, counts + q * KCODES, lpart + (size_t)q * UPD_BLOCKS);
  }

  k_finalize<<<1, 256, 0, stream>>>(counts, lpart, out + (size_t)N_BATCH * C_DIM * T_DIM);
  k_output<<<dim3(T_DIM / 32, C_DIM / 32, N_BATCH), dim3(32, 8), 0, stream>>>(x, resid, out);
}